// TrajectoryModel_38147899523372
// MI455X (gfx1250) — compile-verified
//
#include <hip/hip_runtime.h>
#include <hip/hip_bf16.h>
#include <math.h>

// ---------------- problem constants ----------------
#define BB      128      // batch
#define LL      1024     // sequence length
#define DI      256      // D_INNER
#define DS      64       // D_STATE
#define DTR     8        // DT_RANK
#define NDBC    136      // DT_RANK + 2*D_STATE
#define DM      128      // D_MODEL
#define NHEAD   24       // PRED_LEN*3
#define CHUNK   128      // timesteps per chunk
#define NCHUNK  (LL / CHUNK)
#define RPC     (BB * CHUNK)   // rows per chunk = 16384
#define MTILES  (RPC / 16)     // 1024
#define NTILES  9              // ceil(136/16)

typedef float v2f __attribute__((ext_vector_type(2)));
typedef float v8f __attribute__((ext_vector_type(8)));

// CDNA5 async global->LDS copy (ASYNCcnt-tracked), drained with s_wait_asynccnt.
#define ASYNC_LD_B32(ldsOff, gaddr)                                        \
    asm volatile("global_load_async_to_lds_b32 %0, %1, off"                \
                 :: "v"(ldsOff), "v"(gaddr) : "memory")
#define WAIT_ASYNC0() asm volatile("s_wait_asynccnt 0" ::: "memory")

// ---------------- precompute kernels ----------------

// W_eff[r][j] = sum_m W_inproj[r][m] * W_in[m][j]; b_eff[r] = sum_m W_inproj[r][m]*b_in[m]
__global__ __launch_bounds__(256) void prep_weff_kernel(
    const float* __restrict__ W_in,      // [128,8]
    const float* __restrict__ b_in,      // [128]
    const float* __restrict__ W_inproj,  // [512,128]
    float* __restrict__ weff,            // [512,8]
    float* __restrict__ beff)            // [512]
{
    int r = blockIdx.x * blockDim.x + threadIdx.x;
    if (r >= 2 * DI) return;
    float acc[8];
#pragma unroll
    for (int j = 0; j < 8; ++j) acc[j] = 0.f;
    float bb = 0.f;
    for (int m = 0; m < DM; ++m) {
        float wpm = W_inproj[r * DM + m];
#pragma unroll
        for (int j = 0; j < 8; ++j) acc[j] = fmaf(wpm, W_in[m * 8 + j], acc[j]);
        bb = fmaf(wpm, b_in[m], bb);
    }
#pragma unroll
    for (int j = 0; j < 8; ++j) weff[r * 8 + j] = acc[j];
    beff[r] = bb;
}

__global__ __launch_bounds__(256) void prep_aneg_kernel(
    const float* __restrict__ A_log, float* __restrict__ A_neg)
{
    int i = blockIdx.x * blockDim.x + threadIdx.x;
    if (i < DI * DS) A_neg[i] = -__expf(A_log[i]);
}

// W_final[o][d] = sum_m W_head[o][m] * W_out[m][d]
__global__ __launch_bounds__(256) void prep_wfinal_kernel(
    const float* __restrict__ W_head,   // [24,128]
    const float* __restrict__ W_out,    // [128,256]
    float* __restrict__ W_final)        // [24,256]
{
    int e = blockIdx.x * blockDim.x + threadIdx.x;
    if (e >= NHEAD * DI) return;
    int o = e / DI, d = e % DI;
    float acc = 0.f;
    for (int m = 0; m < DM; ++m)
        acc = fmaf(W_head[o * DM + m], W_out[m * DI + d], acc);
    W_final[o * DI + d] = acc;
}

// ---------------- phase 1: fused inproj + depthwise conv + SiLU ----------------
__global__ __launch_bounds__(256) void conv_kernel(
    const float* __restrict__ x,        // [B,L,8]
    const float* __restrict__ weff,     // [512,8] (rows 0..255 = xc half)
    const float* __restrict__ beff,     // [512]
    const float* __restrict__ conv_w,   // [256,1,4]
    const float* __restrict__ conv_b,   // [256]
    float* __restrict__ xc_out,         // [RPC,256]
    int chunk)
{
    const int b  = blockIdx.x / CHUNK;
    const int tt = blockIdx.x % CHUNK;
    const int t  = chunk * CHUNK + tt;
    const int d  = threadIdx.x;

    __shared__ float xs[4][8];
    if (threadIdx.x < 32) {
        int k = threadIdx.x >> 3, j = threadIdx.x & 7;
        int ts = t - 3 + k;
        xs[k][j] = (ts >= 0) ? x[((size_t)b * LL + ts) * 8 + j] : 0.f;
    }
    __syncthreads();

    float wrow[8];
#pragma unroll
    for (int j = 0; j < 8; ++j) wrow[j] = weff[d * 8 + j];
    const float bias = beff[d];

    float acc = conv_b[d];
#pragma unroll
    for (int k = 0; k < 4; ++k) {
        if (t - 3 + k >= 0) {
            float p = bias;
#pragma unroll
            for (int j = 0; j < 8; ++j) p = fmaf(xs[k][j], wrow[j], p);
            acc = fmaf(conv_w[d * 4 + k], p, acc);
        }
    }
    float s = acc / (1.f + __expf(-acc));   // SiLU
    xc_out[(size_t)(b * CHUNK + tt) * DI + d] = s;
}

// ---------------- phase 2: dbc = xc @ W_xproj.T via f32 WMMA ----------------
// One wave computes a 32(M)x16(N) strip: two 16x16 WMMA tiles sharing the B operand.
__global__ __launch_bounds__(128) void gemm_dbc_kernel(
    const float* __restrict__ xc,       // [RPC,256]
    const float* __restrict__ W_xproj,  // [136,256]
    float* __restrict__ dtlow,          // [RPC,8]
    float* __restrict__ Bm,             // [RPC,64]
    float* __restrict__ Cm)             // [RPC,64]
{
    const int lane = threadIdx.x & 31;
    const int w = blockIdx.x * 4 + (threadIdx.x >> 5);
    const int tile_n  = w % NTILES;
    const int tile_mp = w / NTILES;          // pair of m-tiles
    const int m0 = tile_mp * 32;
    const int n0 = tile_n * 16;

    // A (16x4 f32): lanes 0-15 rows M, K pair (0,1); lanes 16-31 rows M, K pair (2,3)
    // B (4x16 f32): lanes 0-15 cols N, K pair (0,1); lanes 16-31 cols N, K pair (2,3)
    const int ncol = n0 + (lane & 15);
    const int nclamped = (ncol < NDBC - 1) ? ncol : (NDBC - 1);
    const int koff = (lane >> 4) * 2;

    const float* aptr0 = xc + (size_t)(m0 + (lane & 15)) * DI + koff;
    const float* aptr1 = aptr0 + (size_t)16 * DI;
    const float* bptr  = W_xproj + (size_t)nclamped * DI + koff;
    __builtin_prefetch(bptr, 0, 3);

    v8f acc0 = {0.f, 0.f, 0.f, 0.f, 0.f, 0.f, 0.f, 0.f};
    v8f acc1 = acc0;
#pragma unroll 8
    for (int k0 = 0; k0 < DI; k0 += 4) {
        v2f bmat = *(const v2f*)(bptr + k0);
        v2f a0   = *(const v2f*)(aptr0 + k0);
        v2f a1   = *(const v2f*)(aptr1 + k0);
        acc0 = __builtin_amdgcn_wmma_f32_16x16x4_f32(
            false, a0, false, bmat, (short)0, acc0, false, false);
        acc1 = __builtin_amdgcn_wmma_f32_16x16x4_f32(
            false, a1, false, bmat, (short)0, acc1, false, false);
    }

    // D layout: VGPR i -> lanes 0-15: (row m0+i, col n0+lane); lanes 16-31: (row m0+8+i, col n0+lane-16)
    const int nOut  = n0 + (lane & 15);
    const int rbase = m0 + ((lane >> 4) << 3);
    if (nOut < NDBC) {
        // Branch-free epilogue: per-lane destination array + stride, hoisted out of the store loop.
        float* base;
        int stride;
        if (nOut < DTR)            { base = dtlow + nOut;            stride = DTR; }
        else if (nOut < DTR + DS)  { base = Bm + (nOut - DTR);       stride = DS;  }
        else                       { base = Cm + (nOut - DTR - DS);  stride = DS;  }
#pragma unroll
        for (int i = 0; i < 8; ++i)
            base[(size_t)(rbase + i) * stride] = acc0[i];
#pragma unroll
        for (int i = 0; i < 8; ++i)
            base[(size_t)(rbase + 16 + i) * stride] = acc1[i];
    }
}

// ---------------- phase 3: dt = softplus(dtlow @ W_dt.T + b_dt) ----------------
__global__ __launch_bounds__(256) void dt_kernel(
    const float* __restrict__ dtlow,   // [RPC,8]
    const float* __restrict__ W_dt,    // [256,8]
    const float* __restrict__ b_dt,    // [256]
    float* __restrict__ dt_out)        // [RPC,256]
{
    const size_t row = blockIdx.x;
    const int d = threadIdx.x;
    float acc = b_dt[d];
#pragma unroll
    for (int j = 0; j < 8; ++j)
        acc = fmaf(dtlow[row * DTR + j], W_dt[d * DTR + j], acc);
    float sp = (acc > 20.f) ? acc : log1pf(__expf(acc));
    dt_out[row * DI + d] = sp;
}

// ---------------- phase 4: sequential SSM scan ----------------
// One WG per batch, one lane per channel; h[64] + A_row[64] in registers.
// B/C rows staged into double-buffered LDS via async global->LDS copies so the
// next step's data movement overlaps the current step's exp/FMA burst; single
// barrier per timestep.
__global__ __launch_bounds__(256) void scan_kernel(
    const float* __restrict__ dt_ws,   // [RPC,256]
    const float* __restrict__ Bm,      // [RPC,64]
    const float* __restrict__ Cm,      // [RPC,64]
    const float* __restrict__ xc,      // [RPC,256]
    const float* __restrict__ A_neg,   // [256,64]
    float* __restrict__ h_ws,          // [B,256,64]
    const float* __restrict__ x,       // [B,L,8]
    const float* __restrict__ weff,    // [512,8] (rows 256..511 = gate half)
    const float* __restrict__ beff,    // [512]
    const float* __restrict__ Dp,      // [256]
    float* __restrict__ y_gated,       // [B,256]
    int chunk)
{
    const int b = blockIdx.x;
    const int d = threadIdx.x;

    float A_r[DS];
#pragma unroll
    for (int s = 0; s < DS; ++s) A_r[s] = A_neg[d * DS + s];

    float h[DS];
    float* hptr = h_ws + ((size_t)b * DI + d) * DS;
    if (chunk == 0) {
#pragma unroll
        for (int s = 0; s < DS; ++s) h[s] = 0.f;
    } else {
#pragma unroll
        for (int s = 0; s < DS; ++s) h[s] = hptr[s];
    }

    __shared__ float sB[2][DS];
    __shared__ float sC[2][DS];
    const bool last_chunk = (chunk == NCHUNK - 1);
    const size_t row0 = (size_t)b * CHUNK;

    // ---- prime buffer 0 with tt=0 ----
    if (d < DS) {
        unsigned lo = (unsigned)(uintptr_t)(void*)&sB[0][d];
        ASYNC_LD_B32(lo, (const void*)&Bm[row0 * DS + d]);
    } else if (d < 2 * DS) {
        unsigned lo = (unsigned)(uintptr_t)(void*)&sC[0][d - DS];
        ASYNC_LD_B32(lo, (const void*)&Cm[row0 * DS + (d - DS)]);
    }
    float dt_cur = dt_ws[row0 * DI + d];
    float xc_cur = xc[row0 * DI + d];
    WAIT_ASYNC0();
    __syncthreads();

    for (int tt = 0; tt < CHUNK; ++tt) {
        const int cur = tt & 1;
        const int nxt = cur ^ 1;
        const bool have_next = (tt + 1 < CHUNK);
        const size_t rown = row0 + tt + 1;

        // issue next step's B/C copies into the other LDS buffer
        if (have_next) {
            if (d < DS) {
                unsigned lo = (unsigned)(uintptr_t)(void*)&sB[nxt][d];
                ASYNC_LD_B32(lo, (const void*)&Bm[rown * DS + d]);
            } else if (d < 2 * DS) {
                unsigned lo = (unsigned)(uintptr_t)(void*)&sC[nxt][d - DS];
                ASYNC_LD_B32(lo, (const void*)&Cm[rown * DS + (d - DS)]);
            }
        }
        float dt_next = 0.f, xc_next = 0.f;
        if (have_next) {
            dt_next = dt_ws[rown * DI + d];
            xc_next = xc[rown * DI + d];
        }

        const float dtx = dt_cur * xc_cur;
#pragma unroll
        for (int s = 0; s < DS; ++s) {
            float dA = __expf(dt_cur * A_r[s]);
            h[s] = fmaf(h[s], dA, dtx * sB[cur][s]);
        }

        if (last_chunk && tt == CHUNK - 1) {
            float y = 0.f;
#pragma unroll
            for (int s = 0; s < DS; ++s) y = fmaf(h[s], sC[cur][s], y);
            y = fmaf(xc_cur, Dp[d], y);
            // gate z (linear in x) evaluated only at the final timestep
            float z2 = beff[DI + d];
#pragma unroll
            for (int j = 0; j < 8; ++j)
                z2 = fmaf(x[((size_t)b * LL + (LL - 1)) * 8 + j], weff[(DI + d) * 8 + j], z2);
            const float g = z2 / (1.f + __expf(-z2));   // SiLU(z)
            y_gated[b * DI + d] = y * g;
        }

        WAIT_ASYNC0();       // my async writes to LDS landed
        __syncthreads();     // everyone's writes visible; safe to flip buffers
        dt_cur = dt_next;
        xc_cur = xc_next;
    }

    if (!last_chunk) {
#pragma unroll
        for (int s = 0; s < DS; ++s) hptr[s] = h[s];
    }
}

// ---------------- phase 5: head (folded W_head @ W_out) ----------------
__global__ __launch_bounds__(32) void head_kernel(
    const float* __restrict__ y_gated,  // [B,256]
    const float* __restrict__ W_final,  // [24,256]
    const float* __restrict__ b_head,   // [24]
    float* __restrict__ out)            // [B,24]
{
    const int b = blockIdx.x;
    const int o = threadIdx.x;
    if (o >= NHEAD) return;
    float acc = b_head[o];
    for (int dd = 0; dd < DI; ++dd)
        acc = fmaf(y_gated[b * DI + dd], W_final[o * DI + dd], acc);
    out[b * NHEAD + o] = acc;
}

// ---------------- launcher ----------------
extern "C" void kernel_launch(void* const* d_in, const int* in_sizes, int n_in,
                              void* d_out, int out_size, void* d_ws, size_t ws_size,
                              hipStream_t stream) {
    (void)in_sizes; (void)n_in; (void)out_size; (void)ws_size;

    const float* x        = (const float*)d_in[0];
    const float* W_in     = (const float*)d_in[1];
    const float* b_in     = (const float*)d_in[2];
    const float* W_inproj = (const float*)d_in[3];
    const float* conv_w   = (const float*)d_in[4];
    const float* conv_b   = (const float*)d_in[5];
    const float* W_xproj  = (const float*)d_in[6];
    const float* W_dt     = (const float*)d_in[7];
    const float* b_dt     = (const float*)d_in[8];
    const float* A_log    = (const float*)d_in[9];
    const float* Dp       = (const float*)d_in[10];
    const float* W_out    = (const float*)d_in[11];
    const float* W_head   = (const float*)d_in[12];
    const float* b_head   = (const float*)d_in[13];
    float* out = (float*)d_out;

    float* ws = (float*)d_ws;
    float* weff    = ws;  ws += 2 * DI * 8;           // 4096
    float* beff    = ws;  ws += 2 * DI;               // 512
    float* A_neg   = ws;  ws += DI * DS;              // 16384
    float* W_final = ws;  ws += NHEAD * DI;           // 6144
    float* h_ws    = ws;  ws += (size_t)BB * DI * DS; // 2.1M
    float* xc_ws   = ws;  ws += (size_t)RPC * DI;     // 4.2M
    float* dt_ws   = ws;  ws += (size_t)RPC * DI;     // 4.2M
    float* dtlow   = ws;  ws += (size_t)RPC * DTR;    // 131K
    float* Bm_ws   = ws;  ws += (size_t)RPC * DS;     // 1.05M
    float* Cm_ws   = ws;  ws += (size_t)RPC * DS;     // 1.05M
    float* yg      = ws;  ws += BB * DI;              // 32K   (~51 MB total fp32)

    prep_weff_kernel<<<2, 256, 0, stream>>>(W_in, b_in, W_inproj, weff, beff);
    prep_aneg_kernel<<<(DI * DS + 255) / 256, 256, 0, stream>>>(A_log, A_neg);
    prep_wfinal_kernel<<<(NHEAD * DI + 255) / 256, 256, 0, stream>>>(W_head, W_out, W_final);

    for (int c = 0; c < NCHUNK; ++c) {
        conv_kernel<<<RPC, 256, 0, stream>>>(x, weff, beff, conv_w, conv_b, xc_ws, c);
        gemm_dbc_kernel<<<(MTILES / 2) * NTILES / 4, 128, 0, stream>>>(
            xc_ws, W_xproj, dtlow, Bm_ws, Cm_ws);
        dt_kernel<<<RPC, 256, 0, stream>>>(dtlow, W_dt, b_dt, dt_ws);
        scan_kernel<<<BB, 256, 0, stream>>>(dt_ws, Bm_ws, Cm_ws, xc_ws, A_neg, h_ws,
                                            x, weff, beff, Dp, yg, c);
    }
    head_kernel<<<BB, 32, 0, stream>>>(yg, W_final, b_head, out);
}